// TopKGate_80857054315026
// MI455X (gfx1250) — compile-verified
//
#include <hip/hip_runtime.h>
#include <hip/hip_bf16.h>

#define S_TOK 8192
#define HID   4096
#define NEXP  64
#define CAP   256          // ceil(8192 * 2.0 / 64) = 256 (>= MIN_CAPACITY)
#define NBLK  (S_TOK / 16) // 512 gemm tiles

typedef __attribute__((ext_vector_type(2))) float v2f;
typedef __attribute__((ext_vector_type(4))) float v4f;
typedef __attribute__((ext_vector_type(8))) float v8f;

// ---------------------------------------------------------------------------
// 1) Non-temporal zero fill of the entire output (1.07 GB dominates runtime).
// ---------------------------------------------------------------------------
__global__ void zero_out_kernel(float* __restrict__ out, long long n) {
    long long i      = (long long)blockIdx.x * blockDim.x + threadIdx.x;
    long long stride = (long long)gridDim.x * blockDim.x;
    long long n4     = n >> 2;
    v4f z = {0.f, 0.f, 0.f, 0.f};
    v4f* o4 = (v4f*)out;
    for (long long j = i; j < n4; j += stride)
        __builtin_nontemporal_store(z, o4 + j);
    long long t = (n4 << 2) + i;
    if (t < n) out[t] = 0.f;
}

// ---------------------------------------------------------------------------
// 2) Fused router GEMM (fp32 WMMA 16x16x4) + softmax + top-2 per token.
//    One wave per 16-token tile; 4 accumulators cover all 64 experts.
//    A layout (16x4 f32): lanes 0-15 hold K=k0,k0+1 ; lanes 16-31 hold K=k0+2,k0+3
//    B layout (4x16 f32): lane%16 = N column, lane/16 selects K half.
// ---------------------------------------------------------------------------
__global__ __launch_bounds__(32)
void gate_gemm_top2_kernel(const float* __restrict__ X, const float* __restrict__ W,
                           int* __restrict__ sel1, int* __restrict__ sel2,
                           float* __restrict__ g1, float* __restrict__ g2,
                           float* __restrict__ me_part) {
    __shared__ float tile[16][68]; // 16 tokens x 64 experts (padded)

    const int lane  = threadIdx.x;
    const int mrow  = lane & 15;
    const int half  = lane >> 4;
    const int tile0 = blockIdx.x * 16;

    const float* xr  = X + (size_t)(tile0 + mrow) * HID;
    const float* wr0 = W + (size_t)(0  + mrow) * HID;
    const float* wr1 = W + (size_t)(16 + mrow) * HID;
    const float* wr2 = W + (size_t)(32 + mrow) * HID;
    const float* wr3 = W + (size_t)(48 + mrow) * HID;

    v8f acc0 = {}, acc1 = {}, acc2 = {}, acc3 = {};

    #pragma unroll 4
    for (int k0 = 0; k0 < HID; k0 += 4) {
        const int kk = k0 + 2 * half;
        v2f a  = *(const v2f*)(xr  + kk);
        v2f b0 = *(const v2f*)(wr0 + kk);
        v2f b1 = *(const v2f*)(wr1 + kk);
        v2f b2 = *(const v2f*)(wr2 + kk);
        v2f b3 = *(const v2f*)(wr3 + kk);
        acc0 = __builtin_amdgcn_wmma_f32_16x16x4_f32(false, a, false, b0, (short)0, acc0, false, false);
        acc1 = __builtin_amdgcn_wmma_f32_16x16x4_f32(false, a, false, b1, (short)0, acc1, false, false);
        acc2 = __builtin_amdgcn_wmma_f32_16x16x4_f32(false, a, false, b2, (short)0, acc2, false, false);
        acc3 = __builtin_amdgcn_wmma_f32_16x16x4_f32(false, a, false, b3, (short)0, acc3, false, false);
    }

    // C/D layout: VGPR r, lanes 0-15 -> M=r ; lanes 16-31 -> M=8+r ; N = lane%16
    #pragma unroll
    for (int r = 0; r < 8; ++r) {
        const int m = r + 8 * half;
        tile[m][ 0 + mrow] = acc0[r];
        tile[m][16 + mrow] = acc1[r];
        tile[m][32 + mrow] = acc2[r];
        tile[m][48 + mrow] = acc3[r];
    }
    __syncthreads();

    // Per-row softmax + top-2 (lanes 0-15, one token each)
    if (lane < 16) {
        float mx = -3.4e38f;
        for (int e = 0; e < NEXP; ++e) mx = fmaxf(mx, tile[lane][e]);
        float sum = 0.f;
        for (int e = 0; e < NEXP; ++e) sum += __expf(tile[lane][e] - mx);
        const float inv = 1.f / sum;
        float b1v = -1.f, b2v = -1.f; int b1i = 0, b2i = 0;
        for (int e = 0; e < NEXP; ++e) {
            float g = __expf(tile[lane][e] - mx) * inv;
            tile[lane][e] = g; // overwrite logits with gate values
            if (g > b1v)      { b2v = b1v; b2i = b1i; b1v = g; b1i = e; }
            else if (g > b2v) { b2v = g;  b2i = e; }
        }
        const int s = tile0 + lane;
        sel1[s] = b1i; sel2[s] = b2i; g1[s] = b1v; g2[s] = b2v;
    }
    __syncthreads();

    // Deterministic per-block expert gate sums (for l_aux "me" term)
    float s0 = 0.f, s1 = 0.f;
    for (int r = 0; r < 16; ++r) { s0 += tile[r][lane]; s1 += tile[r][lane + 32]; }
    me_part[(size_t)blockIdx.x * 64 + lane]      = s0;
    me_part[(size_t)blockIdx.x * 64 + lane + 32] = s1;
}

// ---------------------------------------------------------------------------
// 3) Ordered capacity-slot scan: one thread per expert (64), chunks staged in
//    LDS so all threads scan tokens in order. Pass 2 continues the counter,
//    matching loc2 = cumsum(mask2)-1 + sum(mask1) exactly.
// ---------------------------------------------------------------------------
__global__ __launch_bounds__(64)
void scan_assign_kernel(const int* __restrict__ sel1, const int* __restrict__ sel2,
                        int* __restrict__ slot1, int* __restrict__ slot2,
                        int* __restrict__ cnt1g) {
    __shared__ int sbuf[256];
    const int t = threadIdx.x; // expert id
    int cnt = 0;
    for (int c = 0; c < S_TOK; c += 256) {
        for (int i = t; i < 256; i += 64) sbuf[i] = sel1[c + i];
        __syncthreads();
        for (int i = 0; i < 256; ++i)
            if (sbuf[i] == t) { int sl = cnt++; slot1[c + i] = (sl < CAP) ? sl : -1; }
        __syncthreads();
    }
    cnt1g[t] = cnt; // pre-drop top1 count per expert (== "ce" numerator)
    for (int c = 0; c < S_TOK; c += 256) {
        for (int i = t; i < 256; i += 64) sbuf[i] = sel2[c + i];
        __syncthreads();
        for (int i = 0; i < 256; ++i)
            if (sbuf[i] == t) { int sl = cnt++; slot2[c + i] = (sl < CAP) ? sl : -1; }
        __syncthreads();
    }
}

// ---------------------------------------------------------------------------
// 4) l_aux = E * sum_e(me[e]*ce[e]) / S^2 ; fixed-order deterministic reduce.
// ---------------------------------------------------------------------------
__global__ __launch_bounds__(64)
void laux_reduce_kernel(const float* __restrict__ me_part, const int* __restrict__ cnt1g,
                        float* __restrict__ out0) {
    __shared__ float red[64];
    const int t = threadIdx.x;
    float gs = 0.f;
    for (int b = 0; b < NBLK; ++b) gs += me_part[(size_t)b * 64 + t];
    red[t] = gs * (float)cnt1g[t];
    __syncthreads();
    if (t == 0) {
        float tot = 0.f;
        for (int e = 0; e < NEXP; ++e) tot += red[e];
        out0[0] = (float)NEXP * tot / ((float)S_TOK * (float)S_TOK);
    }
}

// ---------------------------------------------------------------------------
// 5) Sparse scatter of <=2 nonzeros per token into combine/dispatch tensors.
// ---------------------------------------------------------------------------
__global__ void scatter_combine_kernel(const int* __restrict__ sel1, const int* __restrict__ sel2,
                                       const float* __restrict__ g1, const float* __restrict__ g2,
                                       const int* __restrict__ slot1, const int* __restrict__ slot2,
                                       float* __restrict__ cw, float* __restrict__ dm) {
    const int s = blockIdx.x * blockDim.x + threadIdx.x;
    if (s >= S_TOK) return;
    const int sl1 = slot1[s], sl2 = slot2[s];
    const float w1 = (sl1 >= 0) ? g1[s] : 0.f;
    const float w2 = (sl2 >= 0) ? g2[s] : 0.f;
    const float denom = fmaxf(w1 + w2, 1.1920928955078125e-7f);
    const float w1n = w1 / denom, w2n = w2 / denom;
    const size_t base = (size_t)s * NEXP * CAP;
    if (sl1 >= 0) {
        const size_t off = base + (size_t)sel1[s] * CAP + sl1;
        cw[off] = w1n; dm[off] = (w1n > 0.f) ? 1.f : 0.f;
    }
    if (sl2 >= 0) {
        const size_t off = base + (size_t)sel2[s] * CAP + sl2;
        cw[off] = w2n; dm[off] = (w2n > 0.f) ? 1.f : 0.f;
    }
}

// ---------------------------------------------------------------------------
extern "C" void kernel_launch(void* const* d_in, const int* in_sizes, int n_in,
                              void* d_out, int out_size, void* d_ws, size_t ws_size,
                              hipStream_t stream) {
    const float* X = (const float*)d_in[0]; // [S, H]
    const float* W = (const float*)d_in[1]; // [E, H]
    float* out = (float*)d_out;             // [1 + S*E*C + S*E*C]

    char* w = (char*)d_ws;
    int*   sel1    = (int*)  (w);
    int*   sel2    = (int*)  (w + 4  * (size_t)S_TOK);
    float* g1      = (float*)(w + 8  * (size_t)S_TOK);
    float* g2      = (float*)(w + 12 * (size_t)S_TOK);
    int*   slot1   = (int*)  (w + 16 * (size_t)S_TOK);
    int*   slot2   = (int*)  (w + 20 * (size_t)S_TOK);
    int*   cnt1    = (int*)  (w + 24 * (size_t)S_TOK);         // 64 ints
    float* me_part = (float*)(w + 24 * (size_t)S_TOK + 256);   // [NBLK, 64]

    const size_t sec = (size_t)S_TOK * NEXP * CAP;

    zero_out_kernel<<<4096, 256, 0, stream>>>(out, (long long)out_size);
    gate_gemm_top2_kernel<<<NBLK, 32, 0, stream>>>(X, W, sel1, sel2, g1, g2, me_part);
    scan_assign_kernel<<<1, 64, 0, stream>>>(sel1, sel2, slot1, slot2, cnt1);
    laux_reduce_kernel<<<1, 64, 0, stream>>>(me_part, cnt1, out);
    scatter_combine_kernel<<<(S_TOK + 255) / 256, 256, 0, stream>>>(
        sel1, sel2, g1, g2, slot1, slot2, out + 1, out + 1 + sec);
}